// Net_12799002542605
// MI455X (gfx1250) — compile-verified
//
#include <hip/hip_runtime.h>

typedef __attribute__((ext_vector_type(2))) float v2f;
typedef __attribute__((ext_vector_type(8))) float v8f;

#define NODES 200000
#define EDGES 6400000
#define DIN   128
#define HID   10
#define HP    16      // padded hidden dim (zero cols 10..15)
#define NTILES (NODES / 16)   // 12500, exact

// ---------------- degree / init ----------------

__global__ void k_init(float* __restrict__ dinv_r, float* __restrict__ dinv_c,
                       float* __restrict__ mean16) {
    int i = blockIdx.x * blockDim.x + threadIdx.x;
    if (i < NODES) { dinv_r[i] = 1.0f; dinv_c[i] = 1.0f; }  // self-loop counts
    if (blockIdx.x == 0 && threadIdx.x < 16) mean16[threadIdx.x] = 0.0f;
}

__global__ void k_count(const int* __restrict__ row, const int* __restrict__ col,
                        float* __restrict__ dr, float* __restrict__ dc) {
    int e = blockIdx.x * blockDim.x + threadIdx.x;
    if (e >= EDGES) return;
    unsafeAtomicAdd(&dr[row[e]], 1.0f);
    unsafeAtomicAdd(&dc[col[e]], 1.0f);
}

__global__ void k_rsqrt(float* __restrict__ dr, float* __restrict__ dc) {
    int i = blockIdx.x * blockDim.x + threadIdx.x;
    if (i >= NODES) return;
    dr[i] = rsqrtf(dr[i]);
    dc[i] = rsqrtf(dc[i]);
}

// ---------------- dense transform: Out[N x 16] = A[N x K] @ Wpad[K x 16] ----------------
// One wave per 16-row tile; fp32 WMMA 16x16x4, K/4 chained steps.
// W (wRows x wCols, row-major) is staged into LDS zero-padded to K x 16.

__global__ __launch_bounds__(256)
void k_gemm_wmma(const float* __restrict__ A, const float* __restrict__ W,
                 float* __restrict__ Out, int K, int strideA,
                 int wRows, int wCols, int numTiles) {
    __shared__ float Bs[128 * 16];   // max K = 128
    int tid = threadIdx.x;
    for (int idx = tid; idx < K * 16; idx += blockDim.x) {
        int k = idx >> 4, n = idx & 15;
        Bs[idx] = (k < wRows && n < wCols) ? W[k * wCols + n] : 0.0f;
    }
    __syncthreads();

    int wave = tid >> 5, lane = tid & 31;
    int tile = blockIdx.x * (blockDim.x >> 5) + wave;
    if (tile >= numTiles) return;       // wave-uniform; EXEC stays all-ones for WMMA

    int half = lane >> 4;               // K sub-pair select
    int l    = lane & 15;               // M for A-loads, N for B/C/D
    const float* arow = A + (size_t)(tile * 16 + l) * strideA + 2 * half;

    v8f c = {};
    for (int k0 = 0; k0 < K; k0 += 4) {
        v2f a, b;
        a.x = arow[k0];
        a.y = arow[k0 + 1];
        b.x = Bs[(k0 + 2 * half) * 16 + l];
        b.y = Bs[(k0 + 2 * half + 1) * 16 + l];
        // 8 args: (neg_a, A, neg_b, B, c_mod, C, reuse_a, reuse_b)
        c = __builtin_amdgcn_wmma_f32_16x16x4_f32(false, a, false, b,
                                                  (short)0, c, false, false);
    }

    float* orow = Out + (size_t)tile * 16 * HP;
#pragma unroll
    for (int r = 0; r < 8; ++r) {
        int m = half * 8 + r;           // C/D layout: lanes 0-15 -> M=r, lanes 16-31 -> M=8+r
        orow[m * HP + l] = c[r];
    }
}

// ---------------- GCN aggregation ----------------

// agg[i] = dinv[i]^2 * hpre[i]   (self-loop term; full re-init of agg each call)
__global__ void k_agg_init(const float* __restrict__ hpre, const float* __restrict__ dinv,
                           float* __restrict__ agg) {
    int i = blockIdx.x * blockDim.x + threadIdx.x;
    if (i >= NODES) return;
    float w = dinv[i]; w = w * w;
    const float* h = hpre + (size_t)i * HP;
    float* a = agg + (size_t)i * HP;
#pragma unroll
    for (int t = 0; t < HP; ++t) a[t] = w * h[t];
}

// agg[dst] += dinv[dst]*dinv[src] * hpre[src]
__global__ void k_edge_scatter(const int* __restrict__ src, const int* __restrict__ dst,
                               const float* __restrict__ dinv,
                               const float* __restrict__ hpre, float* __restrict__ agg) {
    int e = blockIdx.x * blockDim.x + threadIdx.x;
    if (e >= EDGES) return;
    int j = src[e], i = dst[e];
    float w = dinv[i] * dinv[j];
    const float* hj = hpre + (size_t)j * HP;
    float* ai = agg + (size_t)i * HP;
#pragma unroll
    for (int t = 0; t < HID; ++t) unsafeAtomicAdd(ai + t, w * hj[t]);
}

// out[i] = relu(agg[i] + b), zero the padding columns
__global__ void k_finalize_relu(const float* __restrict__ agg, const float* __restrict__ b,
                                float* __restrict__ out) {
    int i = blockIdx.x * blockDim.x + threadIdx.x;
    if (i >= NODES) return;
    const float* a = agg + (size_t)i * HP;
    float* o = out + (size_t)i * HP;
#pragma unroll
    for (int t = 0; t < HP; ++t)
        o[t] = (t < HID) ? fmaxf(a[t] + b[t], 0.0f) : 0.0f;
}

// ---------------- output head (10 -> 1 conv) ----------------

__global__ void k_dot_out(const float* __restrict__ h, const float* __restrict__ Wo,
                          float* __restrict__ p) {
    int i = blockIdx.x * blockDim.x + threadIdx.x;
    if (i >= NODES) return;
    const float* hi = h + (size_t)i * HP;
    float s = 0.0f;
#pragma unroll
    for (int t = 0; t < HID; ++t) s += hi[t] * Wo[t];
    p[i] = s;
}

__global__ void k_aggp_init(const float* __restrict__ p, const float* __restrict__ dinv,
                            float* __restrict__ aggp) {
    int i = blockIdx.x * blockDim.x + threadIdx.x;
    if (i >= NODES) return;
    float w = dinv[i];
    aggp[i] = w * w * p[i];
}

__global__ void k_edge_scatter1(const int* __restrict__ src, const int* __restrict__ dst,
                                const float* __restrict__ dinv,
                                const float* __restrict__ p, float* __restrict__ aggp) {
    int e = blockIdx.x * blockDim.x + threadIdx.x;
    if (e >= EDGES) return;
    int j = src[e], i = dst[e];
    unsafeAtomicAdd(&aggp[i], dinv[i] * dinv[j] * p[j]);
}

__global__ void k_probs_final(const float* __restrict__ aggp, const float* __restrict__ bo,
                              float* __restrict__ out) {
    int i = blockIdx.x * blockDim.x + threadIdx.x;
    if (i >= NODES) return;
    out[i] = aggp[i] + bo[0];
}

// ---------------- global mean + scalar heads ----------------

__global__ void k_mean(const float* __restrict__ h, float* __restrict__ mean16) {
    __shared__ float s[16];
    if (threadIdx.x < 16) s[threadIdx.x] = 0.0f;
    __syncthreads();
    int i = blockIdx.x * blockDim.x + threadIdx.x;
    if (i < NODES) {
        const float* hi = h + (size_t)i * HP;
#pragma unroll
        for (int t = 0; t < HID; ++t) atomicAdd(&s[t], hi[t]);  // ds_add_f32
    }
    __syncthreads();
    if (threadIdx.x < HID) unsafeAtomicAdd(&mean16[threadIdx.x], s[threadIdx.x]);
}

__global__ void k_heads(const float* __restrict__ mean16,
                        const float* __restrict__ Wv, const float* __restrict__ bv,
                        const float* __restrict__ Wdn, const float* __restrict__ bdn,
                        float* __restrict__ out) {
    if (threadIdx.x == 0 && blockIdx.x == 0) {
        float v = bv[0], dn = bdn[0];
        const float invN = 1.0f / (float)NODES;
#pragma unroll
        for (int t = 0; t < HID; ++t) {
            float xm = mean16[t] * invN;
            v  += xm * Wv[t];
            dn += xm * Wdn[t];
        }
        out[NODES]     = dn;   // prob_nothing
        out[NODES + 1] = v;    // value
    }
}

// ---------------- launcher ----------------

extern "C" void kernel_launch(void* const* d_in, const int* in_sizes, int n_in,
                              void* d_out, int out_size, void* d_ws, size_t ws_size,
                              hipStream_t stream) {
    const float* x   = (const float*)d_in[0];
    const int*   ei  = (const int*)d_in[1];
    const int*   row = ei;
    const int*   col = ei + EDGES;
    const float* W1  = (const float*)d_in[2];
    const float* b1  = (const float*)d_in[3];
    const float* W2  = (const float*)d_in[4];
    const float* b2  = (const float*)d_in[5];
    const float* Wp  = (const float*)d_in[6];
    const float* bp  = (const float*)d_in[7];
    const float* W3  = (const float*)d_in[8];
    const float* b3  = (const float*)d_in[9];
    const float* Wo  = (const float*)d_in[10];
    const float* bo  = (const float*)d_in[11];
    const float* Wdn = (const float*)d_in[12];
    const float* bdn = (const float*)d_in[13];
    const float* Wv  = (const float*)d_in[14];
    const float* bv  = (const float*)d_in[15];
    float* out = (float*)d_out;

    float* ws = (float*)d_ws;
    float* dinv_r = ws;                       // N
    float* dinv_c = ws + (size_t)NODES;       // N
    float* bufA   = ws + (size_t)2  * NODES;  // N*16 (layer input, zero-padded)
    float* bufPre = ws + (size_t)18 * NODES;  // N*16 (h = A@W)
    float* bufAgg = ws + (size_t)34 * NODES;  // N*16 (aggregation)
    float* p_pre  = ws + (size_t)50 * NODES;  // N
    float* agg_p  = ws + (size_t)51 * NODES;  // N
    float* mean16 = ws + (size_t)52 * NODES;  // 16

    const int TB = 256;
    const int NB_N = (NODES + TB - 1) / TB;
    const int NB_E = (EDGES + TB - 1) / TB;
    const int GB   = (NTILES + 7) / 8;        // 8 waves (tiles) per 256-thread block

    // degrees -> dinv
    k_init  <<<NB_N, TB, 0, stream>>>(dinv_r, dinv_c, mean16);
    k_count <<<NB_E, TB, 0, stream>>>(row, col, dinv_r, dinv_c);
    k_rsqrt <<<NB_N, TB, 0, stream>>>(dinv_r, dinv_c);

    // conv1 (t2s: src=col, dst=row, dinv_r), K=128
    k_gemm_wmma    <<<GB,   TB, 0, stream>>>(x, W1, bufPre, DIN, DIN, DIN, HID, NTILES);
    k_agg_init     <<<NB_N, TB, 0, stream>>>(bufPre, dinv_r, bufAgg);
    k_edge_scatter <<<NB_E, TB, 0, stream>>>(col, row, dinv_r, bufPre, bufAgg);
    k_finalize_relu<<<NB_N, TB, 0, stream>>>(bufAgg, b1, bufA);

    // conv2 (t2s), K=16 padded
    k_gemm_wmma    <<<GB,   TB, 0, stream>>>(bufA, W2, bufPre, HP, HP, HID, HID, NTILES);
    k_agg_init     <<<NB_N, TB, 0, stream>>>(bufPre, dinv_r, bufAgg);
    k_edge_scatter <<<NB_E, TB, 0, stream>>>(col, row, dinv_r, bufPre, bufAgg);
    k_finalize_relu<<<NB_N, TB, 0, stream>>>(bufAgg, b2, bufA);

    // conv_pred1 (s2t: src=row, dst=col, dinv_c)
    k_gemm_wmma    <<<GB,   TB, 0, stream>>>(bufA, Wp, bufPre, HP, HP, HID, HID, NTILES);
    k_agg_init     <<<NB_N, TB, 0, stream>>>(bufPre, dinv_c, bufAgg);
    k_edge_scatter <<<NB_E, TB, 0, stream>>>(row, col, dinv_c, bufPre, bufAgg);
    k_finalize_relu<<<NB_N, TB, 0, stream>>>(bufAgg, bp, bufA);

    // conv3 (t2s)
    k_gemm_wmma    <<<GB,   TB, 0, stream>>>(bufA, W3, bufPre, HP, HP, HID, HID, NTILES);
    k_agg_init     <<<NB_N, TB, 0, stream>>>(bufPre, dinv_r, bufAgg);
    k_edge_scatter <<<NB_E, TB, 0, stream>>>(col, row, dinv_r, bufPre, bufAgg);
    k_finalize_relu<<<NB_N, TB, 0, stream>>>(bufAgg, b3, bufA);   // bufA = h4

    // probs conv (10 -> 1, t2s, no relu)
    k_dot_out      <<<NB_N, TB, 0, stream>>>(bufA, Wo, p_pre);
    k_aggp_init    <<<NB_N, TB, 0, stream>>>(p_pre, dinv_r, agg_p);
    k_edge_scatter1<<<NB_E, TB, 0, stream>>>(col, row, dinv_r, p_pre, agg_p);
    k_probs_final  <<<NB_N, TB, 0, stream>>>(agg_p, bo, out);

    // mean + scalar heads
    k_mean <<<NB_N, TB, 0, stream>>>(bufA, mean16);
    k_heads<<<1, 1, 0, stream>>>(mean16, Wv, bv, Wdn, bdn, out);
}